// SSIMLoss_13331578487280
// MI455X (gfx1250) — compile-verified
//
#include <hip/hip_runtime.h>

// SSIM loss, fused single pass + deterministic reduction.
// img1, img2: (64,1,512,512) f32.  out: scalar f32 = 1 - mean(ssim_map).
//
// Per block (128 threads = 4 wave32): 16-row x 64-col output tile.
//  Phase 1: 22x72 halo tiles of a,b -> LDS.
//           Interior blocks: Tensor Data Mover (tensor_load_to_lds), with
//           right/bottom zero padding from D#.tensor_dim OOB semantics.
//           Edge blocks (x==0 or y==0 halo out of range): scalar zero-padded path.
//  Phase 2: horizontal 7-tap sliding box sums of {a,b,a2,b2,ab} -> LDS (22x64 each)
//  Phase 3: vertical 7-tap box sum as banded matmul via V_WMMA_F32_16X16X4_F32
//  Phase 4: SSIM map + in-block reduction -> per-block partial in d_ws
// Kernel 2: deterministic tree reduction of 16384 partials -> 1 - sum/N.

typedef __attribute__((ext_vector_type(2))) float v2f;
typedef __attribute__((ext_vector_type(8))) float v8f;
typedef __attribute__((ext_vector_type(4))) unsigned int v4u;
typedef __attribute__((ext_vector_type(4))) int v4i;
typedef __attribute__((ext_vector_type(8))) int v8i;

#define IMG_W 512
#define IMG_H 512
#define N_IMG 64
#define TW 64          // output cols per block
#define TH 16          // output rows per block
#define HALO 3
#define LW 72              // loaded cols (70 needed; 72 => dense TDM tile == stride)
#define LH (TH + 2*HALO)   // 22
#define LSTR LW            // LDS stride == TDM tile_dim0 (72 % 64 = 8 banks offset)
#define RSTR 68            // padded LDS stride for row-sum planes (68 % 64 = 4)
#define C1V 0.0001f        // 0.01^2
#define C2V 0.0009f        // 0.03^2

#define HAVE_TDM __has_builtin(__builtin_amdgcn_tensor_load_to_lds)

__device__ __forceinline__ int imin_(int a, int b) { return a < b ? a : b; }

#if HAVE_TDM
// One TDM descriptor: dense LH x LW f32 tile from a 512-stride image into LDS.
// dim0/dim1 = elements remaining in the tensor from the tile origin (per-dim
// OOB reads return zero => free right/bottom halo padding).
__device__ __forceinline__ void tdm_load_tile(const float* gsrc, unsigned lds_off,
                                              unsigned dim0, unsigned dim1) {
  const unsigned long long ga = (unsigned long long)(uintptr_t)gsrc;
  v4u g0;
  g0.x = 1u;                                   // count=1 (valid), flags 0
  g0.y = lds_off;                              // lds_addr (bytes)
  g0.z = (unsigned)(ga & 0xFFFFFFFFull);       // global_addr[31:0]
  g0.w = (unsigned)((ga >> 32) & 0x1FFFFFFull) // global_addr[56:32]
         | (2u << 30);                         // type=2 ("image")
  v8i g1;
  g1[0] = (int)(2u << 16);                     // data_size=2 (4 bytes)
  g1[1] = (int)((dim0 & 0xFFFFu) << 16);       // tensor_dim0[15:0]   @bits[63:48]
  g1[2] = (int)((dim0 >> 16) | ((dim1 & 0xFFFFu) << 16)); // dim0 hi | dim1 lo
  g1[3] = (int)((dim1 >> 16) | ((unsigned)LW << 16));     // dim1 hi | tile_dim0
  g1[4] = (int)LH;                             // tile_dim1 | tile_dim2=0
  g1[5] = IMG_W;                               // tensor_dim0_stride[31:0]
  g1[6] = 0;                                   // stride hi | tensor_dim1_stride lo
  g1[7] = 0;
  const v4i gz = {0, 0, 0, 0};
#if __clang_major__ >= 23
  const v8i gz8 = {0, 0, 0, 0, 0, 0, 0, 0};
  __builtin_amdgcn_tensor_load_to_lds(g0, g1, gz, gz, gz8, 0);
#else
  __builtin_amdgcn_tensor_load_to_lds(g0, g1, gz, gz, 0);
#endif
}
#endif

__global__ __launch_bounds__(128)
void ssim_tile_kernel(const float* __restrict__ A,
                      const float* __restrict__ B,
                      float* __restrict__ partial) {
  __shared__ float sA[LH][LSTR];
  __shared__ float sB[LH][LSTR];
  __shared__ float sRS[5][LH][RSTR];   // row sums: a, b, a2, b2, ab
  __shared__ float sWave[4];

  const int t   = threadIdx.x;
  const int gx0 = blockIdx.x * TW;
  const int gy0 = blockIdx.y * TH;
  const float* __restrict__ Ai = A + (size_t)blockIdx.z * (IMG_W * IMG_H);
  const float* __restrict__ Bi = B + (size_t)blockIdx.z * (IMG_W * IMG_H);

  // ---------- Phase 1: global -> LDS halo tile ----------
#if HAVE_TDM
  // Interior blocks: halo origin is in-bounds on the left/top; right/bottom
  // overhang is zero-filled by TDM's tensor_dim OOB rule.
  const bool interior = (blockIdx.x > 0) && (blockIdx.y > 0);  // block-uniform
  if (interior) {
    const int gxs = gx0 - HALO;            // >= 0 here
    const int gys = gy0 - HALO;            // >= 0 here
    if (t == 0) {
      const unsigned d0 = (unsigned)(IMG_W - gxs);
      const unsigned d1 = (unsigned)(IMG_H - gys);
      const size_t go = (size_t)gys * IMG_W + gxs;
      tdm_load_tile(Ai + go, (unsigned)(uintptr_t)(void*)&sA[0][0], d0, d1);
      tdm_load_tile(Bi + go, (unsigned)(uintptr_t)(void*)&sB[0][0], d0, d1);
    }
#if __has_builtin(__builtin_amdgcn_s_wait_tensorcnt)
    __builtin_amdgcn_s_wait_tensorcnt(0);
#else
    asm volatile("s_wait_tensorcnt 0x0" ::: "memory");
#endif
    asm volatile("" ::: "memory");   // TDM wrote LDS behind the compiler's back
    __syncthreads();
  } else
#endif
  {
    for (int idx = t; idx < LH * LW; idx += 128) {
      const int y  = idx / LW;
      const int x  = idx - y * LW;
      const int gy = gy0 + y - HALO;
      const int gx = gx0 + x - HALO;
      float va = 0.0f, vb = 0.0f;
      if (gy >= 0 && gy < IMG_H && gx >= 0 && gx < IMG_W) {
        const int off = gy * IMG_W + gx;
        va = Ai[off];
        vb = Bi[off];
      }
      sA[y][x] = va;
      sB[y][x] = vb;
    }
    __syncthreads();
  }

  // ---------- Phase 2: horizontal sliding 7-tap sums ----------
  {
    const int row = t >> 2;          // 0..31 (rows >= LH idle)
    const int xq  = (t & 3) << 4;    // 0,16,32,48: 16-col run per thread
    if (row < LH) {
      float sa = 0.f, sb = 0.f, saa = 0.f, sbb = 0.f, sab = 0.f;
      #pragma unroll
      for (int d = 0; d < 7; ++d) {
        const float av = sA[row][xq + d];
        const float bv = sB[row][xq + d];
        sa += av; sb += bv;
        saa = fmaf(av, av, saa);
        sbb = fmaf(bv, bv, sbb);
        sab = fmaf(av, bv, sab);
      }
      int x = xq;
      for (;;) {
        sRS[0][row][x] = sa;
        sRS[1][row][x] = sb;
        sRS[2][row][x] = saa;
        sRS[3][row][x] = sbb;
        sRS[4][row][x] = sab;
        if (x == xq + 15) break;
        const float ao = sA[row][x],     bo = sB[row][x];
        const float an = sA[row][x + 7], bn = sB[row][x + 7];
        sa  += an - ao;
        sb  += bn - bo;
        saa += an * an - ao * ao;
        sbb += bn * bn - bo * bo;
        sab += an * bn - ao * bo;
        ++x;
      }
    }
  }
  __syncthreads();

  // ---------- Phase 3: vertical 7-tap box sum via f32 WMMA ----------
  // out(16x16) = W(16x24, zero-padded band) x rs(24x16), chunked K=4.
  // A layout (16x4 f32): lane M = lane&15; VGPR v holds K = 4j + v + 2*(lane>>4).
  // B layout mirrored: VGPR v, lane l -> K = 4j + v + 2*(l>>4), N = l&15.
  const int lane = t & 31;
  const int wave = t >> 5;             // 0..3 -> 16-col slice
  const int m    = lane & 15;
  const int half = lane >> 4;
  const int bx   = wave * 16 + (lane & 15);

  v8f Cq[5];
  #pragma unroll
  for (int q = 0; q < 5; ++q) {
    v8f acc = {0.f, 0.f, 0.f, 0.f, 0.f, 0.f, 0.f, 0.f};
#if __has_builtin(__builtin_amdgcn_wmma_f32_16x16x4_f32)
    #pragma unroll
    for (int j = 0; j < 6; ++j) {
      const int k0 = 4 * j + 2 * half;
      const int k1 = k0 + 1;
      v2f a, b;
      // band: W[m][k] = 1 iff m <= k <= m+6  (k = 22,23 never in band)
      a.x = (k0 >= m && k0 <= m + 6) ? 1.0f : 0.0f;
      a.y = (k1 >= m && k1 <= m + 6) ? 1.0f : 0.0f;
      // clamp pad rows to a finite value (weight is 0 there)
      b.x = sRS[q][imin_(k0, LH - 1)][bx];
      b.y = sRS[q][imin_(k1, LH - 1)][bx];
      acc = __builtin_amdgcn_wmma_f32_16x16x4_f32(
          false, a, false, b, (short)0, acc, false, false);
    }
#else
    // VALU fallback with identical C-layout mapping: component r of lane l
    // holds pixel (row = r + 8*(l>>4), col = bx).
    #pragma unroll
    for (int r = 0; r < 8; ++r) {
      const int row_out = r + 8 * half;
      float s = 0.f;
      #pragma unroll
      for (int d = 0; d < 7; ++d) s += sRS[q][row_out + d][bx];
      acc[r] = s;
    }
#endif
    Cq[q] = acc;
  }

  // ---------- Phase 4: SSIM map + reduction ----------
  float lsum = 0.f;
  #pragma unroll
  for (int r = 0; r < 8; ++r) {
    const float mu1 = Cq[0][r], mu2 = Cq[1][r];
    const float saa = Cq[2][r], sbb = Cq[3][r], sab = Cq[4][r];
    const float mu11 = mu1 * mu1;
    const float mu22 = mu2 * mu2;
    const float mu12 = mu1 * mu2;
    const float num = (2.f * mu12 + C1V) * (2.f * (sab - mu12) + C2V);
    const float den = (mu11 + mu22 + C1V) * ((saa - mu11) + (sbb - mu22) + C2V);
    lsum += num / den;
  }
  #pragma unroll
  for (int s = 16; s > 0; s >>= 1) lsum += __shfl_xor(lsum, s, 32);
  if (lane == 0) sWave[wave] = lsum;
  __syncthreads();
  if (t == 0) {
    const float tot = sWave[0] + sWave[1] + sWave[2] + sWave[3];
    const int wg = (blockIdx.z * gridDim.y + blockIdx.y) * gridDim.x + blockIdx.x;
    partial[wg] = tot;
  }
}

__global__ __launch_bounds__(256)
void ssim_reduce_kernel(const float* __restrict__ partial, int n,
                        float* __restrict__ out) {
  __shared__ float red[256];
  float s = 0.f;
  for (int i = threadIdx.x; i < n; i += 256) s += partial[i];
  red[threadIdx.x] = s;
  __syncthreads();
  for (int k = 128; k > 0; k >>= 1) {
    if ((int)threadIdx.x < k) red[threadIdx.x] += red[threadIdx.x + k];
    __syncthreads();
  }
  if (threadIdx.x == 0)
    out[0] = 1.0f - red[0] / 16777216.0f;   // N = 64*512*512
}

extern "C" void kernel_launch(void* const* d_in, const int* in_sizes, int n_in,
                              void* d_out, int out_size, void* d_ws, size_t ws_size,
                              hipStream_t stream) {
  const float* img1 = (const float*)d_in[0];
  const float* img2 = (const float*)d_in[1];
  float* out = (float*)d_out;
  float* partial = (float*)d_ws;   // 8*32*64 = 16384 floats = 64 KB scratch

  dim3 grid(IMG_W / TW, IMG_H / TH, N_IMG);   // (8, 32, 64)
  ssim_tile_kernel<<<grid, dim3(128), 0, stream>>>(img1, img2, partial);

  const int nPartial = (IMG_W / TW) * (IMG_H / TH) * N_IMG;
  ssim_reduce_kernel<<<1, dim3(256), 0, stream>>>(partial, nPartial, out);
}